// WeightedSumAggregation_75428215652439
// MI455X (gfx1250) — compile-verified
//
#include <hip/hip_runtime.h>
#include <hip/hip_bf16.h>

// WeightedSumAggregation for MI455X (gfx1250, wave32).
// x:(64,512,8,128) f32 -> out:(64,512,128) f32
// GEMM x@W1 on V_WMMA_F32_16X16X4_F32; softmax + weighted sum on VALU.
// W1 is pre-swizzled in LDS so every WMMA B fragment is one contiguous,
// bank-conflict-free ds_load_b64 with an immediate offset (no repack movs).

typedef float v2f __attribute__((ext_vector_type(2)));
typedef float v8f __attribute__((ext_vector_type(8)));

#define DD   128          // feature dim
#define HH   64           // hidden dim
#define NHEAD 8
#define WAVES 8           // waves per block
#define ROWS_PER_WAVE 16  // 16 rows = 2 (b,s) groups of 8 heads
#define XPAD 132          // padded LDS row stride (floats) -> conflict-free A reads

__global__ __launch_bounds__(256) void wsa_kernel(
    const float* __restrict__ x,
    const float* __restrict__ W1,
    const float* __restrict__ b1,
    const float* __restrict__ W2,
    float* __restrict__ out)
{
    extern __shared__ float lds[];
    float* w1s = lds;                  // 128*64 = 8192 floats, swizzled (see below)
    float* b1s = w1s + DD * HH;        // 64 floats
    float* w2s = b1s + HH;             // 64 floats
    float* xs  = w2s + HH;             // 8 waves * 16 * 132 floats

    const int tid    = threadIdx.x;
    const int lane   = tid & 31;
    const int wave   = tid >> 5;
    const int lane16 = lane & 15;
    const int hi     = lane >> 4;      // 0: lanes 0-15, 1: lanes 16-31

    // ---- stage W1 swizzled into LDS ----
    // Swizzle: element (k,h) -> pos = (c*4+nt)*64 + (hi2*16 + l)*2 + e
    //   c = k>>2, hi2 = (k>>1)&1, e = k&1, nt = h>>4, l = h&15
    // so lane (hi,lane16)'s B fragment for (k0=4c, tile nt) is the contiguous
    // 8-byte pair at w1s + (c*4+nt)*64 + (hi*16+lane16)*2, and for fixed (c,nt)
    // the 32 lanes cover dword banks 0..63 exactly once (conflict-free b64).
    #pragma unroll
    for (int i = tid; i < (DD * HH) / 4; i += 256) {
        float4 v = ((const float4*)W1)[i];
        int k  = i >> 4;           // row of W1 (0..127)
        int h0 = (i & 15) * 4;     // first of 4 consecutive cols
        int c   = k >> 2;
        int hi2 = (k >> 1) & 1;
        int e   = k & 1;
        int nt  = h0 >> 4;         // same tile for all 4 cols (h0 % 16 in {0,4,8,12})
        int l0  = h0 & 15;
        float* dst = w1s + (c * 4 + nt) * 64 + (hi2 * 16 + l0) * 2 + e;
        dst[0] = v.x; dst[2] = v.y; dst[4] = v.z; dst[6] = v.w;
    }
    if (tid < HH)              b1s[tid]      = b1[tid];
    else if (tid < 2 * HH)     w2s[tid - HH] = W2[tid - HH];

    // ---- stage this wave's 16x128 x tile into LDS (coalesced b128) ----
    const int  tileRow0 = (blockIdx.x * WAVES + wave) * ROWS_PER_WAVE; // global row
    const float* xw = x + (size_t)tileRow0 * DD;
    float* xt = xs + wave * ROWS_PER_WAVE * XPAD;
    #pragma unroll
    for (int it = 0; it < 16; ++it) {
        int f   = it * 32 + lane;   // float4 index within tile, 0..511
        int row = f >> 5;           // 32 float4 per row
        int c4  = f & 31;
        float4 v = ((const float4*)xw)[f];
        *((float4*)(xt + row * XPAD + c4 * 4)) = v;
    }
    __syncthreads();

    // ---- phase 1: 16x64 tile of H = x @ W1 via WMMA f32 16x16x4 ----
    v8f acc[4] = {};                      // four 16x16 accum tiles, N=0..63
    const float* abase = xt  + lane16 * XPAD + hi * 2;      // A frag base
    const float* bbase = w1s + (hi * 16 + lane16) * 2;      // B frag base
    #pragma unroll
    for (int c = 0; c < 32; ++c) {        // k0 = 4c
        v2f a = *((const v2f*)(abase + 4 * c));
        #pragma unroll
        for (int nt = 0; nt < 4; ++nt) {
            v2f b = *((const v2f*)(bbase + (c * 4 + nt) * 64));
            acc[nt] = __builtin_amdgcn_wmma_f32_16x16x4_f32(
                false, a, false, b, (short)0, acc[nt], false, false);
        }
    }

    // ---- phase 2: bias + ReLU, h @ W2, half-wave reduction, softmax ----
    // C layout: vgpr e at lane L holds element (M = e + 8*(L>=16), N = L%16)
    float sc[NHEAD];
    #pragma unroll
    for (int e = 0; e < NHEAD; ++e) sc[e] = 0.0f;
    #pragma unroll
    for (int nt = 0; nt < 4; ++nt) {
        float bias = b1s[nt * 16 + lane16];
        float w2v  = w2s[nt * 16 + lane16];
        #pragma unroll
        for (int e = 0; e < NHEAD; ++e) {
            float hval = fmaxf(acc[nt][e] + bias, 0.0f);
            sc[e] = fmaf(hval, w2v, sc[e]);
        }
    }
    // reduce over N within each 16-lane half (rows M=e resp. M=e+8)
    #pragma unroll
    for (int m = 1; m < 16; m <<= 1) {
        #pragma unroll
        for (int e = 0; e < NHEAD; ++e)
            sc[e] += __shfl_xor(sc[e], m, 32);
    }
    // lanes 0-15: scores of group g0 (heads 0..7); lanes 16-31: group g1.
    // b2 is a scalar added to every score -> softmax invariant, dropped exactly.
    float mx = sc[0];
    #pragma unroll
    for (int e = 1; e < NHEAD; ++e) mx = fmaxf(mx, sc[e]);
    float wv[NHEAD];
    float wsum = 0.0f;
    #pragma unroll
    for (int e = 0; e < NHEAD; ++e) { wv[e] = __expf(sc[e] - mx); wsum += wv[e]; }
    float inv = __frcp_rn(wsum);
    #pragma unroll
    for (int e = 0; e < NHEAD; ++e) wv[e] *= inv;

    // ---- phase 3: out[g,d] = sum_h wv[h] * x[g,h,d] from the LDS tile ----
    const float* xg = xt + hi * NHEAD * XPAD;   // rows 0-7 (g0) or 8-15 (g1)
    float o[8] = {0,0,0,0,0,0,0,0};
    #pragma unroll
    for (int h = 0; h < NHEAD; ++h) {
        const float* xr = xg + h * XPAD + lane16 * 8;
        float4 p0 = *((const float4*)(xr));
        float4 p1 = *((const float4*)(xr + 4));
        float w = wv[h];
        o[0] = fmaf(w, p0.x, o[0]); o[1] = fmaf(w, p0.y, o[1]);
        o[2] = fmaf(w, p0.z, o[2]); o[3] = fmaf(w, p0.w, o[3]);
        o[4] = fmaf(w, p1.x, o[4]); o[5] = fmaf(w, p1.y, o[5]);
        o[6] = fmaf(w, p1.z, o[6]); o[7] = fmaf(w, p1.w, o[7]);
    }
    const size_t gidx = (size_t)(tileRow0 / NHEAD) + hi;  // group index
    float* og = out + gidx * DD + lane16 * 8;
    *((float4*)(og))     = make_float4(o[0], o[1], o[2], o[3]);
    *((float4*)(og + 4)) = make_float4(o[4], o[5], o[6], o[7]);
}

extern "C" void kernel_launch(void* const* d_in, const int* in_sizes, int n_in,
                              void* d_out, int out_size, void* d_ws, size_t ws_size,
                              hipStream_t stream) {
    const float* x  = (const float*)d_in[0];   // (64,512,8,128)
    const float* W1 = (const float*)d_in[1];   // (128,64)
    const float* b1 = (const float*)d_in[2];   // (64,)
    const float* W2 = (const float*)d_in[3];   // (64,1)
    // d_in[4] = b2: scalar shift of all scores -> softmax invariant, unused.
    float* out = (float*)d_out;                // (64,512,128)

    const int groups = 64 * 512;                           // 32768
    const int groupsPerBlock = WAVES * 2;                  // 16
    const int grid = groups / groupsPerBlock;              // 2048
    const size_t shmem =
        (size_t)(DD * HH + HH + HH + WAVES * ROWS_PER_WAVE * XPAD) * sizeof(float); // ~100.9 KB

    wsa_kernel<<<grid, 256, shmem, stream>>>(x, W1, b1, W2, out);
}